// ChebNet_1357209665946
// MI455X (gfx1250) — compile-verified
//
#include <hip/hip_runtime.h>
#include <hip/hip_bf16.h>

typedef float v2f __attribute__((ext_vector_type(2)));
typedef float v8f __attribute__((ext_vector_type(8)));

__device__ __forceinline__ void atomic_add_f32(float* p, float v) {
    // lowers to global_atomic_add_f32 on gfx1250 (no CAS loop)
    unsafeAtomicAdd(p, v);
}

// ---------------- graph normalization ----------------

__global__ __launch_bounds__(256) void k_deg(const int* __restrict__ src,
                                             const float* __restrict__ w,
                                             float* __restrict__ deg, int E) {
    int e = blockIdx.x * blockDim.x + threadIdx.x;
    if (e < E) atomic_add_f32(&deg[src[e]], w[e]);
}

// deg -> dis (in place), and diag
__global__ __launch_bounds__(256) void k_node_norm(float* __restrict__ deg_dis,
                                                   float* __restrict__ diag, int N) {
    int i = blockIdx.x * blockDim.x + threadIdx.x;
    if (i >= N) return;
    float d = deg_dis[i];
    bool pos = d > 0.0f;
    deg_dis[i] = pos ? rsqrtf(fmaxf(d, 1e-12f)) : 0.0f;
    diag[i]    = pos ? 0.0f : -1.0f;
}

__global__ __launch_bounds__(256) void k_edge_norm(const int* __restrict__ src,
                                                   const int* __restrict__ dst,
                                                   const float* __restrict__ w,
                                                   const float* __restrict__ dis,
                                                   float* __restrict__ wn, int E) {
    int e = blockIdx.x * blockDim.x + threadIdx.x;
    if (e < E) wn[e] = -dis[src[e]] * w[e] * dis[dst[e]];
}

// ---------------- small utility kernels ----------------

// out[i*ldo+f] = in[i*ldi+f]   (place x into slab 0 of Z1)
__global__ __launch_bounds__(256) void k_copy(const float* __restrict__ in, int ldi,
                                              float* __restrict__ out, int ldo,
                                              int N, int F) {
    int t = blockIdx.x * blockDim.x + threadIdx.x;
    if (t >= N * F) return;
    int i = t / F, f = t % F;
    out[(size_t)i * ldo + f] = in[(size_t)i * ldi + f];
}

// zero-pad columns: out[r*ldo+c] = (c < Cin) ? in[r*ldi+c] : 0
__global__ __launch_bounds__(256) void k_pad(const float* __restrict__ in, int ldi,
                                             float* __restrict__ out, int ldo,
                                             int R, int Cin) {
    int t = blockIdx.x * blockDim.x + threadIdx.x;
    if (t >= R * ldo) return;
    int r = t / ldo, c = t % ldo;
    out[t] = (c < Cin) ? in[(size_t)r * ldi + c] : 0.0f;
}

// out[i*ldo+f] = scale*diag[i]*p1[i*ld1+f] + beta*p2[i*ld2+f]
__global__ __launch_bounds__(256) void k_init(const float* __restrict__ p1, int ld1,
                                              const float* __restrict__ p2, int ld2,
                                              const float* __restrict__ diag,
                                              float* __restrict__ out, int ldo,
                                              int N, int F, float scale, float beta) {
    int t = blockIdx.x * blockDim.x + threadIdx.x;
    if (t >= N * F) return;
    int i = t / F, f = t % F;
    float v = scale * diag[i] * p1[(size_t)i * ld1 + f]
            + beta * p2[(size_t)i * ld2 + f];
    out[(size_t)i * ldo + f] = v;
}

// edge-parallel scatter: out[dst] += scale*wn[e]*hin[src], F4 float4-chunks per row.
// F=128 -> 32 chunks = one wave per edge, fully coalesced 512B gather per edge.
__global__ __launch_bounds__(256) void k_scatter(const int* __restrict__ src,
                                                 const int* __restrict__ dst,
                                                 const float* __restrict__ wn,
                                                 const float* __restrict__ hin, int ldi4,
                                                 float* __restrict__ out, int ldo4,
                                                 int E, int F4, float scale) {
    long long t = (long long)blockIdx.x * blockDim.x + threadIdx.x;
    if (t >= (long long)E * F4) return;
    int e = (int)(t / F4);
    int c = (int)(t % F4);
    float wv = wn[e] * scale;
    int s = src[e], d = dst[e];
    const float4 hv = *((const float4*)(hin + (size_t)s * (ldi4 * 4)) + c);
    float* op = out + (size_t)d * (ldo4 * 4) + (size_t)c * 4;
    atomic_add_f32(op + 0, wv * hv.x);
    atomic_add_f32(op + 1, wv * hv.y);
    atomic_add_f32(op + 2, wv * hv.z);
    atomic_add_f32(op + 3, wv * hv.w);
}

// ---------------- fp32 WMMA GEMM: out = Z @ W + bias (optional relu) ----------------
// One wave computes a 16-row x (NT*16)-col strip. KDIM/LDW/NT compile-time so the
// inner loop is branch-free with immediate-offset B loads; A fragment reused NT times.
// W and bias must be loadable for all NT*16 columns (padded); stores guarded by Ncols.
template <int KDIM, int LDW, int NT, int RELU>
__global__ __launch_bounds__(256) void k_gemm_wmma(const float* __restrict__ Z, int ldz,
                                                   const float* __restrict__ W,
                                                   const float* __restrict__ bias,
                                                   float* __restrict__ out, int ldo,
                                                   int Mtiles, int Ncols) {
    int wave = (int)((blockIdx.x * (long long)blockDim.x + threadIdx.x) >> 5);
    if (wave >= Mtiles) return;            // wave-uniform: whole wave exits

    int lane = threadIdx.x & 31;
    int lm   = lane & 15;
    int hi   = lane >> 4;                  // 0: K pair {0,1}, 1: K pair {2,3}

    int rowA = wave * 16 + lm;             // A-matrix row for this lane
    const float* arow  = Z + (size_t)rowA * ldz + 2 * hi;   // Z[rowA][k+2hi]
    const float* wbase = W + (size_t)(2 * hi) * LDW + lm;   // W[k+2hi][nt*16+lm]

    v8f acc[NT];
#pragma unroll
    for (int nt = 0; nt < NT; ++nt) acc[nt] = (v8f){};

#pragma unroll 8
    for (int k = 0; k < KDIM; k += 4) {
        v2f a = *(const v2f*)(arow + k);
        const float* wk = wbase + (size_t)k * LDW;
#pragma unroll
        for (int nt = 0; nt < NT; ++nt) {
            v2f b;
            b.x = wk[nt * 16];             // W[k+2hi  ][nt*16+lm]
            b.y = wk[nt * 16 + LDW];       // W[k+2hi+1][nt*16+lm]
            acc[nt] = __builtin_amdgcn_wmma_f32_16x16x4_f32(
                /*neg_a=*/false, a, /*neg_b=*/false, b,
                /*c_mod=*/(short)0, acc[nt], /*reuse_a=*/false, /*reuse_b=*/false);
        }
    }

#pragma unroll
    for (int nt = 0; nt < NT; ++nt) {
        int col = nt * 16 + lm;
        float bv = bias[col];              // bias padded to NT*16, always valid
        bool cOK = col < Ncols;
#pragma unroll
        for (int v = 0; v < 8; ++v) {
            int r = wave * 16 + v + 8 * hi;        // D layout: VGPR v -> row v+8*hi
            float val = acc[nt][v] + bv;
            if (RELU) val = fmaxf(val, 0.0f);
            if (cOK) out[(size_t)r * ldo + col] = val;
        }
    }
}

// ---------------- launcher ----------------

extern "C" void kernel_launch(void* const* d_in, const int* in_sizes, int n_in,
                              void* d_out, int out_size, void* d_ws, size_t ws_size,
                              hipStream_t stream) {
    (void)in_sizes; (void)n_in; (void)out_size; (void)ws_size;
    const int N = 50000, E = 1600000, F = 128, H = 64, C = 40;
    const int CP = 48;                         // C padded to tile multiple

    const float* x  = (const float*)d_in[0];
    const int*   ei = (const int*)d_in[1];
    const float* ew = (const float*)d_in[2];
    const float* W1 = (const float*)d_in[3];
    const float* b1 = (const float*)d_in[4];
    const float* W2 = (const float*)d_in[5];
    const float* b2 = (const float*)d_in[6];
    const int* src = ei;          // edge_index[0,:]
    const int* dst = ei + E;      // edge_index[1,:]
    float* outp = (float*)d_out;

    char* ws = (char*)d_ws;
    size_t off = 0;
    auto alloc = [&](size_t bytes) -> void* {
        off = (off + 255) & ~(size_t)255;
        void* p = ws + off;
        off += bytes;
        return p;
    };
    float* deg  = (float*)alloc((size_t)N * 4);            // becomes dis in place
    float* diag = (float*)alloc((size_t)N * 4);
    float* wn   = (float*)alloc((size_t)E * 4);
    float* W2p  = (float*)alloc((size_t)(4 * H) * CP * 4); // [256, 48] padded W2
    float* b2p  = (float*)alloc((size_t)CP * 4);           // [48] padded b2
    float* Z1   = (float*)alloc((size_t)N * 4 * F * 4);    // [N, 512] f32
    float* Z2   = (float*)alloc((size_t)N * 4 * H * 4);    // [N, 256] f32

    const int T = 256;
    auto cdiv = [](long long a, long long b) { return (int)((a + b - 1) / b); };

    // ----- normalization + weight padding -----
    hipMemsetAsync(deg, 0, (size_t)N * 4, stream);
    k_deg<<<cdiv(E, T), T, 0, stream>>>(src, ew, deg, E);
    k_node_norm<<<cdiv(N, T), T, 0, stream>>>(deg, diag, N);
    k_edge_norm<<<cdiv(E, T), T, 0, stream>>>(src, dst, ew, deg, wn, E);
    k_pad<<<cdiv((long long)4 * H * CP, T), T, 0, stream>>>(W2, C, W2p, CP, 4 * H, C);
    k_pad<<<cdiv(CP, T), T, 0, stream>>>(b2, C, b2p, CP, 1, C);

    // ----- layer 1: Chebyshev terms into Z1 slabs (ld = 512) -----
    const int LD1 = 4 * F;                                  // 512
    k_copy<<<cdiv((long long)N * F, T), T, 0, stream>>>(x, F, Z1, LD1, N, F);
    // T1 = L_hat @ T0
    k_init<<<cdiv((long long)N * F, T), T, 0, stream>>>(Z1, LD1, Z1, LD1, diag,
                                                        Z1 + F, LD1, N, F, 1.0f, 0.0f);
    k_scatter<<<cdiv((long long)E * (F / 4), T), T, 0, stream>>>(src, dst, wn,
        Z1, LD1 / 4, Z1 + F, LD1 / 4, E, F / 4, 1.0f);
    // T2 = 2*L_hat@T1 - T0
    k_init<<<cdiv((long long)N * F, T), T, 0, stream>>>(Z1 + F, LD1, Z1, LD1, diag,
                                                        Z1 + 2 * F, LD1, N, F, 2.0f, -1.0f);
    k_scatter<<<cdiv((long long)E * (F / 4), T), T, 0, stream>>>(src, dst, wn,
        Z1 + F, LD1 / 4, Z1 + 2 * F, LD1 / 4, E, F / 4, 2.0f);
    // T3 = 2*L_hat@T2 - T1
    k_init<<<cdiv((long long)N * F, T), T, 0, stream>>>(Z1 + 2 * F, LD1, Z1 + F, LD1, diag,
                                                        Z1 + 3 * F, LD1, N, F, 2.0f, -1.0f);
    k_scatter<<<cdiv((long long)E * (F / 4), T), T, 0, stream>>>(src, dst, wn,
        Z1 + 2 * F, LD1 / 4, Z1 + 3 * F, LD1 / 4, E, F / 4, 2.0f);

    // h = relu(Z1 @ W1 + b1) -> Z2 slab 0 (ld = 256); W1 is [512,64], no padding needed
    const int LD2 = 4 * H;                                  // 256
    {
        int Mt = N / 16;                                    // 3125 waves
        k_gemm_wmma<512, 64, 4, 1><<<cdiv(Mt, T / 32), T, 0, stream>>>(
            Z1, LD1, W1, b1, Z2, LD2, Mt, H);
    }

    // ----- layer 2: Chebyshev terms into Z2 slabs -----
    k_init<<<cdiv((long long)N * H, T), T, 0, stream>>>(Z2, LD2, Z2, LD2, diag,
                                                        Z2 + H, LD2, N, H, 1.0f, 0.0f);
    k_scatter<<<cdiv((long long)E * (H / 4), T), T, 0, stream>>>(src, dst, wn,
        Z2, LD2 / 4, Z2 + H, LD2 / 4, E, H / 4, 1.0f);
    k_init<<<cdiv((long long)N * H, T), T, 0, stream>>>(Z2 + H, LD2, Z2, LD2, diag,
                                                        Z2 + 2 * H, LD2, N, H, 2.0f, -1.0f);
    k_scatter<<<cdiv((long long)E * (H / 4), T), T, 0, stream>>>(src, dst, wn,
        Z2 + H, LD2 / 4, Z2 + 2 * H, LD2 / 4, E, H / 4, 2.0f);
    k_init<<<cdiv((long long)N * H, T), T, 0, stream>>>(Z2 + 2 * H, LD2, Z2 + H, LD2, diag,
                                                        Z2 + 3 * H, LD2, N, H, 2.0f, -1.0f);
    k_scatter<<<cdiv((long long)E * (H / 4), T), T, 0, stream>>>(src, dst, wn,
        Z2 + 2 * H, LD2 / 4, Z2 + 3 * H, LD2 / 4, E, H / 4, 2.0f);

    // out = Z2 @ W2p + b2p -> d_out [N, 40] (stores guarded at C=40, no relu)
    {
        int Mt = N / 16;                                    // 3125 waves
        k_gemm_wmma<256, 48, 3, 0><<<cdiv(Mt, T / 32), T, 0, stream>>>(
            Z2, LD2, W2p, b2p, outp, C, Mt, C);
    }
}